// Tag2Encodeout_40991167873467
// MI455X (gfx1250) — compile-verified
//
#include <hip/hip_runtime.h>

// CDNA5 / gfx1250, wave32. Matrix ops via V_WMMA_F32_16X16X4_F32 (fp32 exact,
// matching the fp32 reference; workload is bound by the 512 MiB score write,
// so bf16 downconversion buys nothing on the roofline).

typedef __attribute__((ext_vector_type(2))) float v2f;
typedef __attribute__((ext_vector_type(8))) float v8f;

#define NB 32
#define TQ 2048
#define TK 2048
#define TAG_DIM 128   // query feature dim == att dim used in the bmm
#define ENC_DIM 256
#define ATT_DIM 128

// ---------------------------------------------------------------------------
// Zero the colsum accumulator (re-zeroed every launch -> deterministic work).
__global__ void zero_colsum_kernel(float* __restrict__ colsum) {
    int i = blockIdx.x * blockDim.x + threadIdx.x;
    if (i < NB * TK) colsum[i] = 0.0f;
}

// ---------------------------------------------------------------------------
// proj = src @ W^T + bias.  src: [NB*TK, ENC_DIM], W: [ATT_DIM, ENC_DIM],
// dst: [NB*TK, ATT_DIM].  One workgroup = 16 rows; wave w owns d-tile w (8
// tiles of 16 cover ATT_DIM=128). 64 WMMA f32 16x16x4 per wave.
__global__ __launch_bounds__(256) void proj_kernel(const float* __restrict__ src,
                                                   const float* __restrict__ W,
                                                   const float* __restrict__ bias,
                                                   float* __restrict__ dst) {
    __shared__ float s_a[16 * ENC_DIM];  // 16 KB: 16 source rows
    const int rowbase = blockIdx.x * 16;
    const int tid = threadIdx.x;

    // Coalesced stage of 16 contiguous rows (16*256 floats).
    {
        const float4* g = (const float4*)(src + (size_t)rowbase * ENC_DIM);
        float4* s4 = (float4*)s_a;
        for (int i = tid; i < 16 * ENC_DIM / 4; i += 256) s4[i] = g[i];
    }
    __syncthreads();

    const int wave = tid >> 5, lane = tid & 31;
    const int m  = lane & 15;          // row of A / col (N) of B per ISA layout
    const int kb = (lane >> 4) * 2;    // K sub-offset: lanes 0-15 -> K0,1; 16-31 -> K2,3
    const int dbase = wave * 16;

    v8f acc = {0.f, 0.f, 0.f, 0.f, 0.f, 0.f, 0.f, 0.f};
    const float* arow = s_a + m * ENC_DIM;
    const float* wrow = W + (size_t)(dbase + m) * ENC_DIM;  // B[kk][n] = W[dbase+n][e+kk]

#pragma unroll
    for (int e = 0; e < ENC_DIM; e += 4) {
        v2f a = *(const v2f*)(arow + e + kb);
        v2f b = *(const v2f*)(wrow + e + kb);
        acc = __builtin_amdgcn_wmma_f32_16x16x4_f32(false, a, false, b,
                                                    (short)0, acc, false, false);
    }

    // D layout: VGPR r, lane l -> row = r + 8*(l>>4), col = l&15.
    const float bval = bias[dbase + m];
    const int rb = (lane >> 4) * 8;
    float* dcol = dst + (size_t)(rowbase + rb) * ATT_DIM + dbase + m;
#pragma unroll
    for (int r = 0; r < 8; ++r) dcol[(size_t)r * ATT_DIM] = acc[r] + bval;
}

// ---------------------------------------------------------------------------
// Attention: one workgroup per (batch, 16-row q tile).
//   dot[16 x 2048] = query_tile @ kproj[b]^T   (WMMA f32 16x16x4, K=128)
//   softmax rows in LDS, write score, accumulate colsum via fp32 atomics.
__global__ __launch_bounds__(256) void attn_kernel(const float* __restrict__ query,
                                                   const float* __restrict__ kproj,
                                                   float* __restrict__ score,
                                                   float* __restrict__ colsum) {
    __shared__ float s_q[16 * TAG_DIM];  // 8 KB
    __shared__ float s_s[16 * TK];       // 128 KB score strip (total 136 KB LDS)

    const int b = blockIdx.x >> 7;        // / (TQ/16)
    const int qt = blockIdx.x & 127;
    const int qbase = qt * 16;
    const int tid = threadIdx.x;

    // Stage the 16x128 query tile (contiguous rows) into LDS.
    {
        const float4* g = (const float4*)(query + ((size_t)b * TQ + qbase) * TAG_DIM);
        float4* s4 = (float4*)s_q;
        for (int i = tid; i < 16 * TAG_DIM / 4; i += 256) s4[i] = g[i];
    }
    __syncthreads();

    const int wave = tid >> 5, lane = tid & 31;
    const int m  = lane & 15;
    const int kb = (lane >> 4) * 2;
    const float* qrow = s_q + m * TAG_DIM;

    // Each wave computes 16 of the 128 k-tiles (uniform trip count: EXEC all-1s).
    for (int t = wave; t < TK / 16; t += 8) {
        const float* krow = kproj + ((size_t)b * TK + t * 16 + m) * ATT_DIM;
        v8f acc = {0.f, 0.f, 0.f, 0.f, 0.f, 0.f, 0.f, 0.f};
#pragma unroll
        for (int d = 0; d < TAG_DIM; d += 4) {
            v2f a = *(const v2f*)(qrow + d + kb);   // A: query rows (LDS)
            v2f bk = *(const v2f*)(krow + d + kb);  // B: kproj rows (global/L2)
            acc = __builtin_amdgcn_wmma_f32_16x16x4_f32(false, a, false, bk,
                                                        (short)0, acc, false, false);
        }
        const int col = t * 16 + m;
        const int rb = (lane >> 4) * 8;
#pragma unroll
        for (int r = 0; r < 8; ++r) s_s[(rb + r) * TK + col] = acc[r];
    }
    __syncthreads();

    // Softmax: wave w handles rows 2w and 2w+1; wave32 shuffle reductions.
#pragma unroll
    for (int rr = 0; rr < 2; ++rr) {
        const int r = wave * 2 + rr;
        float* row = s_s + r * TK;

        float mx = -3.402823466e38f;
        for (int i = lane; i < TK; i += 32) mx = fmaxf(mx, row[i]);
#pragma unroll
        for (int off = 16; off > 0; off >>= 1) mx = fmaxf(mx, __shfl_xor(mx, off, 32));

        float sum = 0.f;
        for (int i = lane; i < TK; i += 32) {
            float e = __expf(row[i] - mx);
            row[i] = e;
            sum += e;
        }
#pragma unroll
        for (int off = 16; off > 0; off >>= 1) sum += __shfl_xor(sum, off, 32);

        const float inv = 1.0f / sum;
        float* out = score + ((size_t)b * TQ + qbase + r) * TK;
        for (int i = lane; i < TK; i += 32) {
            float p = row[i] * inv;
            row[i] = p;        // keep normalized value for colsum pass
            out[i] = p;        // coalesced 512 MiB score write (the roofline floor)
        }
    }
    __syncthreads();

    // Per-workgroup column sums over the 16 q rows -> global fp32 atomics.
    for (int k = tid; k < TK; k += 256) {
        float s = 0.f;
#pragma unroll
        for (int r = 0; r < 16; ++r) s += s_s[r * TK + k];
        atomicAdd(&colsum[b * TK + k], s);
    }
}

// ---------------------------------------------------------------------------
// att[b,d] = (1/TQ) * sum_k colsum[b,k] * vproj[b,k,d]
__global__ void out_kernel(const float* __restrict__ colsum,
                           const float* __restrict__ vproj,
                           float* __restrict__ att) {
    const int b = blockIdx.x;
    const int d = threadIdx.x;
    const float* cs = colsum + b * TK;
    const float* vp = vproj + (size_t)b * TK * ATT_DIM + d;
    float acc = 0.f;
    for (int k = 0; k < TK; ++k) acc += cs[k] * vp[(size_t)k * ATT_DIM];
    att[b * ATT_DIM + d] = acc * (1.0f / (float)TQ);
}

// ---------------------------------------------------------------------------
extern "C" void kernel_launch(void* const* d_in, const int* in_sizes, int n_in,
                              void* d_out, int out_size, void* d_ws, size_t ws_size,
                              hipStream_t stream) {
    const float* query = (const float*)d_in[0];
    const float* key   = (const float*)d_in[1];
    const float* value = (const float*)d_in[2];
    // d_in[3] = Wq, d_in[4] = bq: dead values in the reference (q is unused).
    const float* Wk = (const float*)d_in[5];
    const float* bk = (const float*)d_in[6];
    const float* Wv = (const float*)d_in[7];
    const float* bv = (const float*)d_in[8];

    float* out   = (float*)d_out;
    float* att   = out;                         // [32, 128]
    float* score = out + NB * ATT_DIM;          // [32, 2048, 2048]

    float* ws     = (float*)d_ws;
    float* kproj  = ws;                                   // 32 MiB
    float* vproj  = ws + (size_t)NB * TK * ATT_DIM;       // 32 MiB
    float* colsum = vproj + (size_t)NB * TK * ATT_DIM;    // 256 KB

    zero_colsum_kernel<<<(NB * TK + 255) / 256, 256, 0, stream>>>(colsum);
    proj_kernel<<<NB * TK / 16, 256, 0, stream>>>(key, Wk, bk, kproj);
    proj_kernel<<<NB * TK / 16, 256, 0, stream>>>(value, Wv, bv, vproj);
    attn_kernel<<<NB * (TQ / 16), 256, 0, stream>>>(query, kproj, score, colsum);
    out_kernel<<<NB, ATT_DIM, 0, stream>>>(colsum, vproj, att);
}